// GDMLTorchPredict_27642409517428
// MI455X (gfx1250) — compile-verified
//
#include <hip/hip_runtime.h>
#include <math.h>

// ---- problem constants (from reference) ----
#define N_ATOMS 21
#define DESC    210          // N_ATOMS*(N_ATOMS-1)/2
#define DPAD    224          // DESC padded to multiple of 16 (14 WMMA tiles)
#define MCONF   64
#define NTRAIN  8192
#define QCONST  0.22360679774997896f   // sqrt(5)/SIG, SIG=10
#define KC      (5.0f / 300.0f)        // 5/(3*SIG^2)
#define STD_C   1.0f
#define C_OFF   0.0f

// k2 tiling: 8 waves/block, each owns one 16x16 (m,t) tile
#define K2_WAVES 8
// k3 split-K: 16 chunks of 512 t's, staged through LDS 128 at a time
#define TCHUNK  512
#define TSTAGE  128
#define WSTRIDE 130          // 128 + 2 pad: conflict-free b64 LDS reads
#define ASTRIDE 226          // 224 + 2 pad: conflict-free b64 LDS reads

typedef float v8f __attribute__((ext_vector_type(8)));
typedef float v2f __attribute__((ext_vector_type(2)));

// ---- workspace layout (float offsets), total ~18.2 MB ----
#define OFF_AQ     0                               // [64][224]  A = q*xs
#define OFF_ANORM  (OFF_AQ    + MCONF*DPAD)        // [64]       ||A[m]||^2
#define OFF_S1     (OFF_ANORM + MCONF)             // [64]       sum_t w1
#define OFF_ES     (OFF_S1    + MCONF)             // [64]       sum_t w2*dot
#define OFF_BN2    (OFF_ES    + MCONF)             // [8192]     ||B[t]||^2
#define OFF_CT     (OFF_BN2   + NTRAIN)            // [8192]     B[t]·J[t]
#define OFF_FSX    (OFF_CT    + NTRAIN)            // [64][224]  Fs_x
#define OFF_BT     (OFF_FSX   + MCONF*DPAD)        // [224][8192] (q*xs_train)^T
#define OFF_JT     (OFF_BT    + DPAD*NTRAIN)       // [224][8192] J^T
#define OFF_W1     (OFF_JT    + DPAD*NTRAIN)       // [64][8192]
#define OFF_W2     (OFF_W1    + MCONF*NTRAIN)      // [64][8192]

// D = A x B + C, fp32 16x16x4 WMMA (codegen-confirmed last round)
static __device__ __forceinline__ v8f wmma_f32(v2f a, v2f b, v8f c) {
  return __builtin_amdgcn_wmma_f32_16x16x4_f32(false, a, false, b, (short)0, c,
                                               false, false);
}

// ---- Phase 0: descriptors A[m,d] = q/dist, ||A[m]||^2, zero accumulators ----
__global__ __launch_bounds__(256) void k0_desc(const float* __restrict__ Rs,
                                               float* __restrict__ ws) {
  const int m = blockIdx.x, tid = threadIdx.x;
  float* Aq = ws + OFF_AQ;
  __shared__ float red[256];
  float sq = 0.f;
  if (tid < DPAD) {
    float v = 0.f;
    if (tid < DESC) {
      // invert row-major lower-tri index: d = i*(i-1)/2 + j, i>j
      int i = 1;
      while ((i + 1) * i / 2 <= tid) ++i;
      const int j = tid - i * (i - 1) / 2;
      const float* ri = Rs + (size_t)(m * N_ATOMS + i) * 3;
      const float* rj = Rs + (size_t)(m * N_ATOMS + j) * 3;
      const float dx = ri[0] - rj[0], dy = ri[1] - rj[1], dz = ri[2] - rj[2];
      v = QCONST / sqrtf(dx * dx + dy * dy + dz * dz);
    }
    Aq[m * DPAD + tid] = v;
    sq = v * v;
  }
  red[tid] = sq;
  __syncthreads();
  for (int s = 128; s > 0; s >>= 1) {
    if (tid < s) red[tid] += red[tid + s];
    __syncthreads();
  }
  if (tid == 0) ws[OFF_ANORM + m] = red[0];
  if (m == 0 && tid < MCONF) { ws[OFF_S1 + tid] = 0.f; ws[OFF_ES + tid] = 0.f; }
}

// ---- Phase 1: transpose q*xs_train, J to [DPAD][NTRAIN]; per-t norms/dots ----
__global__ __launch_bounds__(256) void k1_prep(const float* __restrict__ xs,
                                               const float* __restrict__ J,
                                               float* __restrict__ ws) {
  const int t = blockIdx.x * 256 + threadIdx.x;   // [0, 8192)
  float* BT = ws + OFF_BT;
  float* JT = ws + OFF_JT;
  float n2 = 0.f, cc = 0.f;
  for (int d = 0; d < DESC; ++d) {
    const float b = QCONST * xs[(size_t)t * DESC + d];   // per-lane sequential row walk
    const float jv = J[(size_t)t * DESC + d];
    BT[(size_t)d * NTRAIN + t] = b;                      // coalesced across lanes
    JT[(size_t)d * NTRAIN + t] = jv;
    n2 += b * b;
    cc += b * jv;
  }
  for (int d = DESC; d < DPAD; ++d) {                    // zero pad
    BT[(size_t)d * NTRAIN + t] = 0.f;
    JT[(size_t)d * NTRAIN + t] = 0.f;
  }
  ws[OFF_BN2 + t] = n2;
  ws[OFF_CT + t] = cc;
}

// ---- Phase 2: G1 = A·B^T, G2 = A·J^T (64x8192, K=210) + distance/exp epilogue
// 8 waves/block share one LDS A-tile; each wave owns a 16x16 (m,t) tile.
__global__ __launch_bounds__(32 * K2_WAVES) void k2_gemm1(float* __restrict__ ws) {
  const int m0 = blockIdx.x * 16;
  const int tid = threadIdx.x;
  const int lane = tid & 31, wv = tid >> 5;
  const int t0 = (blockIdx.y * K2_WAVES + wv) * 16;
  const int half = lane >> 4;       // which half of the wave
  const int l15 = lane & 15;        // row (A frag) / col (B,C frags)
  const float* Aq = ws + OFF_AQ;
  const float* BT = ws + OFF_BT;
  const float* JT = ws + OFF_JT;

  __shared__ float As[16 * ASTRIDE];                // padded: conflict-free reads
  for (int k = tid; k < 16 * DPAD; k += 32 * K2_WAVES)
    As[(k / DPAD) * ASTRIDE + (k % DPAD)] = Aq[m0 * DPAD + k];
  __syncthreads();

  v8f acc1 = {}, acc2 = {};
  for (int d = 0; d < DPAD; d += 4) {
    const int ks = d + 2 * half;
    v2f a;                                   // A 16x4: lanes 0-15 K=0,1; 16-31 K=2,3
    a.x = As[l15 * ASTRIDE + ks];
    a.y = As[l15 * ASTRIDE + ks + 1];
    const size_t b0 = (size_t)ks * NTRAIN + t0 + l15;
    const size_t b1 = (size_t)(ks + 1) * NTRAIN + t0 + l15;
    v2f bB, bJ;                              // B 4x16 mirrored layout, coalesced
    bB.x = BT[b0]; bB.y = BT[b1];
    bJ.x = JT[b0]; bJ.y = JT[b1];
    acc1 = wmma_f32(a, bB, acc1);
    acc2 = wmma_f32(a, bJ, acc2);
  }

  union { v8f v; float f[8]; } U1, U2;
  U1.v = acc1; U2.v = acc2;
  const float bn2 = ws[OFF_BN2 + t0 + l15];
  const float ct  = ws[OFF_CT + t0 + l15];
  float* w1 = ws + OFF_W1;
  float* w2 = ws + OFF_W2;
  for (int r = 0; r < 8; ++r) {
    const int m = m0 + r + 8 * half;         // C/D layout: VGPR r -> rows r / r+8
    const int t = t0 + l15;
    const float an2 = ws[OFF_ANORM + m];
    const float d2 = an2 + bn2 - 2.f * U1.f[r];
    const float xd = sqrtf(fmaxf(d2, 0.f));
    const float e = KC * __expf(-xd);
    const float dotv = U2.f[r] - ct;
    const float w1v = e * dotv;              // exp_xs * dot
    const float w2v = e * (1.f + xd);        // exp_xs_1
    w1[(size_t)m * NTRAIN + t] = w1v;
    w2[(size_t)m * NTRAIN + t] = w2v;
    float s1v = w1v, esv = w2v * dotv;       // reduce over the 16 t-cols (half-wave)
    for (int off = 1; off < 16; off <<= 1) {
      s1v += __shfl_xor(s1v, off, 32);
      esv += __shfl_xor(esv, off, 32);
    }
    if (l15 == 0) {
      atomicAdd(&ws[OFF_S1 + m], s1v);
      atomicAdd(&ws[OFF_ES + m], esv);
    }
  }
}

// ---- Phase 2b: FsX := A * S1 * STD  (split-K partials of phase 3 add onto it)
__global__ __launch_bounds__(256) void k2b_init_fsx(float* __restrict__ ws) {
  const int m = blockIdx.x, tid = threadIdx.x;
  if (tid < DPAD) {
    ws[OFF_FSX + m * DPAD + tid] =
        ws[OFF_AQ + m * DPAD + tid] * ws[OFF_S1 + m] * STD_C;
  }
}

// ---- Phase 3: P1 = W1·xs_train, P2 = W2·J (64x210, split-K over 16 t-chunks)
// FsX -= (q*P1 + P2) * STD via f32 atomics. w-tiles staged through padded LDS.
__global__ __launch_bounds__(32) void k3_gemm2(const float* __restrict__ xs,
                                               const float* __restrict__ J,
                                               float* __restrict__ ws) {
  const int m0 = blockIdx.x * 16;
  const int d0 = blockIdx.y * 16;
  const int tc0 = blockIdx.z * TCHUNK;
  const int lane = threadIdx.x;
  const int half = lane >> 4;
  const int l15 = lane & 15;
  const float* w1 = ws + OFF_W1;
  const float* w2 = ws + OFF_W2;
  const int dcol = d0 + l15;
  const bool dok = dcol < DESC;

  __shared__ float W1s[16 * WSTRIDE];
  __shared__ float W2s[16 * WSTRIDE];

  v8f p1 = {}, p2 = {};
  for (int tc = tc0; tc < tc0 + TCHUNK; tc += TSTAGE) {
    __syncthreads();
    for (int k = lane; k < 16 * TSTAGE; k += 32) {       // coalesced 128-float rows
      const int row = k >> 7, col = k & (TSTAGE - 1);
      W1s[row * WSTRIDE + col] = w1[(size_t)(m0 + row) * NTRAIN + tc + col];
      W2s[row * WSTRIDE + col] = w2[(size_t)(m0 + row) * NTRAIN + tc + col];
    }
    __syncthreads();
    for (int tt = 0; tt < TSTAGE; tt += 4) {
      const int ts = tt + 2 * half;
      v2f a1, a2;                                        // conflict-free ds b64
      a1.x = W1s[l15 * WSTRIDE + ts]; a1.y = W1s[l15 * WSTRIDE + ts + 1];
      a2.x = W2s[l15 * WSTRIDE + ts]; a2.y = W2s[l15 * WSTRIDE + ts + 1];
      const int tg = tc + ts;
      v2f b1, b2;                                        // row-major, coalesced in d
      b1.x = dok ? xs[(size_t)tg * DESC + dcol] : 0.f;
      b1.y = dok ? xs[(size_t)(tg + 1) * DESC + dcol] : 0.f;
      b2.x = dok ? J[(size_t)tg * DESC + dcol] : 0.f;
      b2.y = dok ? J[(size_t)(tg + 1) * DESC + dcol] : 0.f;
      p1 = wmma_f32(a1, b1, p1);
      p2 = wmma_f32(a2, b2, p2);
    }
  }

  union { v8f v; float f[8]; } U1, U2;
  U1.v = p1; U2.v = p2;
  float* FsX = ws + OFF_FSX;
  if (dok) {
    for (int r = 0; r < 8; ++r) {
      const int m = m0 + r + 8 * half;
      atomicAdd(&FsX[m * DPAD + dcol],
                (-QCONST * U1.f[r] - U2.f[r]) * STD_C);
    }
  }
}

// ---- Phase 4: scatter to forces Fs[m,a,:] = sum_b FsX/r^3 * (r_a - r_b); Es ----
__global__ __launch_bounds__(32) void k4_final(const float* __restrict__ Rs,
                                               const float* __restrict__ ws,
                                               float* __restrict__ out) {
  const int m = blockIdx.x;
  const int a = threadIdx.x;
  if (a == 31) out[m] = C_OFF + STD_C * ws[OFF_ES + m] / QCONST;  // Es
  if (a >= N_ATOMS) return;
  const float* R = Rs + (size_t)m * N_ATOMS * 3;
  const float ax = R[a * 3 + 0], ay = R[a * 3 + 1], az = R[a * 3 + 2];
  const float* FsX = ws + OFF_FSX + (size_t)m * DPAD;
  float fx = 0.f, fy = 0.f, fz = 0.f;
  for (int b = 0; b < N_ATOMS; ++b) {
    if (b == a) continue;
    const float dx = ax - R[b * 3 + 0];
    const float dy = ay - R[b * 3 + 1];
    const float dz = az - R[b * 3 + 2];
    const float r2 = dx * dx + dy * dy + dz * dz;
    const float rinv = rsqrtf(r2);
    const int i = a > b ? a : b;
    const int j = a > b ? b : a;
    const float w = FsX[i * (i - 1) / 2 + j] * rinv * rinv * rinv;
    fx += w * dx; fy += w * dy; fz += w * dz;
  }
  float* Fo = out + MCONF + (size_t)(m * N_ATOMS + a) * 3;
  Fo[0] = fx; Fo[1] = fy; Fo[2] = fz;
}

extern "C" void kernel_launch(void* const* d_in, const int* in_sizes, int n_in,
                              void* d_out, int out_size, void* d_ws, size_t ws_size,
                              hipStream_t stream) {
  const float* Rs = (const float*)d_in[0];   // [64,21,3]
  const float* xs = (const float*)d_in[1];   // [8192,210]
  const float* J  = (const float*)d_in[2];   // [8192,210]
  float* ws = (float*)d_ws;
  float* out = (float*)d_out;                // [64] Es ++ [64,21,3] Fs

  k0_desc    <<<MCONF, 256, 0, stream>>>(Rs, ws);
  k1_prep    <<<NTRAIN / 256, 256, 0, stream>>>(xs, J, ws);
  k2_gemm1   <<<dim3(MCONF / 16, (NTRAIN / 16) / K2_WAVES), 32 * K2_WAVES, 0,
                stream>>>(ws);
  k2b_init_fsx<<<MCONF, 256, 0, stream>>>(ws);
  k3_gemm2   <<<dim3(MCONF / 16, DPAD / 16, NTRAIN / TCHUNK), 32, 0, stream>>>(
                xs, J, ws);
  k4_final   <<<MCONF, 32, 0, stream>>>(Rs, ws, out);
}